// OurCustomGraphLayer_55731495633081
// MI455X (gfx1250) — compile-verified
//
#include <hip/hip_runtime.h>

// ---------------------------------------------------------------------------
// Batched graph aggregation on MI455X (gfx1250, wave32):
//   A = edges + I;  out = nan0((A @ nodes) / rowsum(A))
//
// Identity hoisted out of the GEMM:
//   A @ X     = edges @ X + X
//   rowsum(A) = rowsum(edges) + 1
// so the 512-iteration hot loop streams raw edges (non-temporal b64 loads,
// 256 MiB read once) and runs pure WMMA; +I applied in epilogue.
//
// f32 WMMA 16x16x4.  Wave = 32 rows x 64 cols (2 strips x 4 d-tiles,
// 8 x v8f accumulators) so each B fragment feeds 2 WMMAs:
//   per k-step: 2 A-b64(NT) + 8 B-b32 loads, 8 v_wmma, 2 v_pk_add_f32.
// Row-sums kept as packed v2f accumulators so v_pk_add_f32 reads the A
// fragment register pair in place (no v_mov shuffling in the loop).
// Block = 128 thr (4 waves) = 128 rows; grid = (2048/128, B) = (16,16).
// ---------------------------------------------------------------------------

typedef __attribute__((ext_vector_type(2))) float v2f;
typedef __attribute__((ext_vector_type(8))) float v8f;

constexpr int kN = 2048;           // nodes per graph
constexpr int kD = 64;             // feature dim
constexpr int kRowsPerWave  = 32;  // two 16-row WMMA strips
constexpr int kWavesPerBlock = 4;
constexpr int kRowsPerBlock = kRowsPerWave * kWavesPerBlock;  // 128

__global__ __launch_bounds__(kWavesPerBlock * 32)
void gcn_aggregate_wmma(const float* __restrict__ nodes,
                        const float* __restrict__ edges,
                        float* __restrict__ out) {
  const int tile = blockIdx.x;            // row-tile within graph
  const int b    = blockIdx.y;            // batch index
  const int wave = threadIdx.x >> 5;
  const int lane = threadIdx.x & 31;
  const int m    = lane & 15;             // row-in-strip (A) / col (B,D)
  const int hi   = lane >> 4;             // 0: K={0,1}; 1: K={2,3}

  const int i0 = tile * kRowsPerBlock + wave * kRowsPerWave;  // first row

  // A-row pointers for the two 16-row strips (this lane's rows)
  const float* __restrict__ A0 = edges + ((size_t)b * kN + (i0 + m)) * kN;
  const float* __restrict__ A1 = A0 + (size_t)16 * kN;
  const float* __restrict__ X  = nodes + (size_t)b * kN * kD;

  v8f acc[2][4] = {};                     // [strip][d-tile]
  v2f rsv0 = {0.0f, 0.0f};                // packed row-sum partials
  v2f rsv1 = {0.0f, 0.0f};

#pragma unroll 4
  for (int k0 = 0; k0 < kN; k0 += 4) {
    const int ka = k0 + 2 * hi;           // even -> 8B aligned v2f load

    // A fragments: one 8-byte non-temporal load per strip (streamed once)
    v2f a0 = __builtin_nontemporal_load((const v2f*)(A0 + ka));
    v2f a1 = __builtin_nontemporal_load((const v2f*)(A1 + ka));
    rsv0 += a0;                           // v_pk_add_f32, in-place operands
    rsv1 += a1;

    // B fragments (4x16 f32) per 16-wide d-tile; nodes is L2-resident
    const float* __restrict__ Xr0 = X + (size_t)ka * kD;  // row ka
    const float* __restrict__ Xr1 = Xr0 + kD;             // row ka+1
    v2f bf[4];
#pragma unroll
    for (int t = 0; t < 4; ++t) {
      bf[t].x = Xr0[m + 16 * t];
      bf[t].y = Xr1[m + 16 * t];
    }

    // 8 WMMAs: each B fragment reused by both row strips
#pragma unroll
    for (int t = 0; t < 4; ++t) {
      acc[0][t] = __builtin_amdgcn_wmma_f32_16x16x4_f32(
          false, a0, false, bf[t], (short)0, acc[0][t], false, false);
      acc[1][t] = __builtin_amdgcn_wmma_f32_16x16x4_f32(
          false, a1, false, bf[t], (short)0, acc[1][t], false, false);
    }
  }

  // Horizontal reduce of packed partials, then cross-half combine:
  // lanes l and l+16 hold complementary K-halves of row (l&15)'s sum.
  float rs0 = rsv0.x + rsv0.y;
  float rs1 = rsv1.x + rsv1.y;
  rs0 += __shfl_xor(rs0, 16, 32);
  rs1 += __shfl_xor(rs1, 16, 32);
  rs0 += 1.0f;                            // +I contribution to rowsum
  rs1 += 1.0f;

  // Epilogue: num += X[row] (the +I term), divide by denom, nan->0, store.
#pragma unroll
  for (int s = 0; s < 2; ++s) {
    const float rs = (s == 0) ? rs0 : rs1;
#pragma unroll
    for (int r = 0; r < 8; ++r) {
      const int orow = r + 8 * hi;        // C/D layout: VGPR r -> row r+8*hi
      const int g    = i0 + s * 16 + orow;            // global row
      const float inv = 1.0f / __shfl(rs, orow, 32);  // denom for that row

      const float* __restrict__ xrow = X + (size_t)g * kD;
      float* __restrict__ o = out + ((size_t)b * kN + g) * kD + m;
#pragma unroll
      for (int t = 0; t < 4; ++t) {
        float v = (acc[s][t][r] + xrow[16 * t + m]) * inv;
        v = (v != v) ? 0.0f : v;          // tf.where(isnan, 0, x)
        o[16 * t] = v;
      }
    }
  }
}

extern "C" void kernel_launch(void* const* d_in, const int* in_sizes, int n_in,
                              void* d_out, int out_size, void* d_ws, size_t ws_size,
                              hipStream_t stream) {
  (void)n_in; (void)out_size; (void)d_ws; (void)ws_size;
  const float* nodes = (const float*)d_in[0];  // (B, N, D) f32
  const float* edges = (const float*)d_in[1];  // (B, N, N) f32
  float* out = (float*)d_out;                  // (B, N, D) f32

  const int B = in_sizes[0] / (kN * kD);       // 16
  dim3 grid(kN / kRowsPerBlock, B);            // (16, 16) blocks
  gcn_aggregate_wmma<<<grid, kWavesPerBlock * 32, 0, stream>>>(nodes, edges, out);
}